// GAT_15659450761218
// MI455X (gfx1250) — compile-verified
//
#include <hip/hip_runtime.h>
#include <hip/hip_bf16.h>

#define HEADS 3
#define CPH   128
#define F     384      // HEADS*CPH
#define HIN   12

typedef __attribute__((ext_vector_type(2))) float v2f;
typedef __attribute__((ext_vector_type(8))) float v8f;

__device__ __forceinline__ unsigned flipf(float f) {
  unsigned b = __float_as_uint(f);
  return (b & 0x80000000u) ? ~b : (b | 0x80000000u);
}
__device__ __forceinline__ float unflipf(unsigned u) {
  unsigned b = (u & 0x80000000u) ? (u & 0x7FFFFFFFu) : ~u;
  return __uint_as_float(b);
}
__device__ __forceinline__ float lrelu(float v) { return v > 0.f ? v : 0.2f * v; }

// ---------------- zero fill (AGG + DEN + MAX contiguous region) ----------------
__global__ void k_zero(float* __restrict__ p, unsigned long long n) {
  unsigned long long i = (unsigned long long)blockIdx.x * blockDim.x + threadIdx.x;
  if (i < n) p[i] = 0.0f;
}

// ------------- layer-1 projection (K=12) + attention dots, 1 block/node -------------
__global__ void k_proj1(const float* __restrict__ x, const float* __restrict__ W1,
                        const float* __restrict__ a_src, const float* __restrict__ a_dst,
                        float* __restrict__ xp, float* __restrict__ asrc,
                        float* __restrict__ adst) {
  const int n = blockIdx.x, t = threadIdx.x;   // blockDim = 128
  __shared__ float xrow[HIN];
  __shared__ float red[6][CPH];
  if (t < HIN) xrow[t] = x[(size_t)n * HIN + t];
  __syncthreads();
#pragma unroll
  for (int h = 0; h < HEADS; ++h) {
    const int c = h * CPH + t;
    float acc = 0.f;
#pragma unroll
    for (int k = 0; k < HIN; ++k) acc = fmaf(xrow[k], W1[k * F + c], acc);
    xp[(size_t)n * F + c] = acc;
    red[h][t]     = acc * a_src[c];
    red[3 + h][t] = acc * a_dst[c];
  }
  __syncthreads();
  for (int off = CPH / 2; off > 0; off >>= 1) {
    if (t < off) {
#pragma unroll
      for (int j = 0; j < 6; ++j) red[j][t] += red[j][t + off];
    }
    __syncthreads();
  }
  if (t < HEADS)          asrc[(size_t)n * HEADS + t]          = red[t][0];
  else if (t < 2 * HEADS) adst[(size_t)n * HEADS + (t - HEADS)] = red[t][0];
}

// ------------- attention dots from existing features (layer 2), 1 block/node -------------
__global__ void k_alpha(const float* __restrict__ xp, const float* __restrict__ a_src,
                        const float* __restrict__ a_dst, float* __restrict__ asrc,
                        float* __restrict__ adst) {
  const int n = blockIdx.x, t = threadIdx.x;
  __shared__ float red[6][CPH];
#pragma unroll
  for (int h = 0; h < HEADS; ++h) {
    const int c = h * CPH + t;
    const float v = xp[(size_t)n * F + c];
    red[h][t]     = v * a_src[c];
    red[3 + h][t] = v * a_dst[c];
  }
  __syncthreads();
  for (int off = CPH / 2; off > 0; off >>= 1) {
    if (t < off) {
#pragma unroll
      for (int j = 0; j < 6; ++j) red[j][t] += red[j][t + off];
    }
    __syncthreads();
  }
  if (t < HEADS)          asrc[(size_t)n * HEADS + t]          = red[t][0];
  else if (t < 2 * HEADS) adst[(size_t)n * HEADS + (t - HEADS)] = red[t][0];
}

// ------------- edge pass a: segment max via uint-flipped atomicMax -------------
__global__ void k_edge_max(const int* __restrict__ src, const int* __restrict__ dst,
                           int E, int N, const float* __restrict__ asrc,
                           const float* __restrict__ adst, unsigned* __restrict__ mx) {
  const int i = blockIdx.x * blockDim.x + threadIdx.x;
  if (i >= E + N) return;
  int s, d;
  if (i < E) { s = src[i]; d = dst[i]; } else { s = i - E; d = s; }
#pragma unroll
  for (int h = 0; h < HEADS; ++h) {
    float e = lrelu(asrc[(size_t)s * HEADS + h] + adst[(size_t)d * HEADS + h]);
    atomicMax(&mx[(size_t)d * HEADS + h], flipf(e));
  }
}

// ------------- edge pass b: exp(e - max), store per-edge, atomic denom -------------
__global__ void k_edge_expsum(const int* __restrict__ src, const int* __restrict__ dst,
                              int E, int N, const float* __restrict__ asrc,
                              const float* __restrict__ adst, const unsigned* __restrict__ mx,
                              float* __restrict__ ex, float* __restrict__ den) {
  const int i = blockIdx.x * blockDim.x + threadIdx.x;
  if (i >= E + N) return;
  int s, d;
  if (i < E) { s = src[i]; d = dst[i]; } else { s = i - E; d = s; }
#pragma unroll
  for (int h = 0; h < HEADS; ++h) {
    float e = lrelu(asrc[(size_t)s * HEADS + h] + adst[(size_t)d * HEADS + h]);
    float v = expf(e - unflipf(mx[(size_t)d * HEADS + h]));
    ex[(size_t)i * HEADS + h] = v;
    atomicAdd(&den[(size_t)d * HEADS + h], v);
  }
}

// ------------- edge pass c: weighted scatter-add of 384-wide rows, 1 block/edge -------------
__global__ void k_edge_agg(const int* __restrict__ src, const int* __restrict__ dst,
                           int E, int N, const float* __restrict__ ex,
                           const float* __restrict__ den, const float* __restrict__ xp,
                           float* __restrict__ agg) {
  const int i = blockIdx.x, t = threadIdx.x;   // blockDim = 128
  int s, d;
  if (i < E) { s = src[i]; d = dst[i]; } else { s = i - E; d = s; }
#pragma unroll
  for (int h = 0; h < HEADS; ++h) {
    const float alpha = ex[(size_t)i * HEADS + h] / den[(size_t)d * HEADS + h];
    const int c = h * CPH + t;
    atomicAdd(&agg[(size_t)d * F + c], alpha * xp[(size_t)s * F + c]);
  }
}

// ------------- rowwise bias add: agg[n][c] += b[c] -------------
__global__ void k_bias_rows(float* __restrict__ agg, const float* __restrict__ b,
                            unsigned long long total) {
  unsigned long long i = (unsigned long long)blockIdx.x * blockDim.x + threadIdx.x;
  if (i < total) agg[i] += b[i % F];
}

// ------------- layer-2 GEMM: C[M,384] = A[M,384] @ B[384,384], fp32 WMMA -------------
// One wave computes a 16x32 output strip (two 16x16 tiles, shared A operand).
// A 16x4 fp32 layout: lane l -> M = l&15, K = (l>>4)*2 .. +1 (2 VGPRs).
// B 4x16 fp32 layout: lane l -> N = l&15, K = (l>>4)*2 .. +1 (2 VGPRs).
// C/D: VGPR i -> M = i + ((lane>=16)?8:0), N = lane&15.
__global__ void __launch_bounds__(256) k_gemm(const float* __restrict__ A,
                                              const float* __restrict__ B,
                                              float* __restrict__ C, int mtiles) {
  const int lane = threadIdx.x & 31;
  const int wave = threadIdx.x >> 5;
  const int wt = blockIdx.x * 8 + wave;       // wave-tile id
  const int npair = F / 32;                   // 12 strips of 32 columns
  const int tm = wt / npair;
  const int tp = wt % npair;
  if (tm >= mtiles) return;                   // wave-uniform: EXEC stays all-1s
  const int m  = tm * 16 + (lane & 15);
  const int n0 = tp * 32 + (lane & 15);
  const int n1 = n0 + 16;
  const int kb = (lane >> 4) * 2;
  const float* arow = A + (size_t)m * F;
  v8f acc0 = {};
  v8f acc1 = {};
  for (int k0 = 0; k0 < F; k0 += 4) {
    const int k = k0 + kb;
    v2f a;  a.x  = arow[k];            a.y  = arow[k + 1];
    const float* bp = B + (size_t)k * F;
    v2f b0; b0.x = bp[n0];             b0.y = bp[F + n0];
    v2f b1; b1.x = bp[n1];             b1.y = bp[F + n1];
    acc0 = __builtin_amdgcn_wmma_f32_16x16x4_f32(false, a, false, b0, (short)0, acc0,
                                                 false, false);
    acc1 = __builtin_amdgcn_wmma_f32_16x16x4_f32(false, a, false, b1, (short)0, acc1,
                                                 false, false);
  }
  const int mr = tm * 16 + ((lane >> 4) << 3);
#pragma unroll
  for (int i = 0; i < 8; ++i) {
    C[(size_t)(mr + i) * F + n0] = acc0[i];
    C[(size_t)(mr + i) * F + n1] = acc1[i];
  }
}

// ------------- final: out[n] = dot(agg[n] + b2, Wl) + bl -------------
__global__ void k_final(const float* __restrict__ agg, const float* __restrict__ b2,
                        const float* __restrict__ Wl, const float* __restrict__ bl,
                        float* __restrict__ out) {
  const int n = blockIdx.x, t = threadIdx.x;   // blockDim = 128
  __shared__ float red[CPH];
  float acc = 0.f;
#pragma unroll
  for (int h = 0; h < HEADS; ++h) {
    const int c = h * CPH + t;
    acc = fmaf(agg[(size_t)n * F + c] + b2[c], Wl[c], acc);
  }
  red[t] = acc;
  __syncthreads();
  for (int off = CPH / 2; off > 0; off >>= 1) {
    if (t < off) red[t] += red[t + off];
    __syncthreads();
  }
  if (t == 0) out[n] = red[0] + bl[0];
}

extern "C" void kernel_launch(void* const* d_in, const int* in_sizes, int n_in,
                              void* d_out, int out_size, void* d_ws, size_t ws_size,
                              hipStream_t stream) {
  const float* x   = (const float*)d_in[0];
  const int*   ei  = (const int*)d_in[1];
  const float* W1  = (const float*)d_in[2];
  const float* as1 = (const float*)d_in[3];
  const float* ad1 = (const float*)d_in[4];
  const float* b1  = (const float*)d_in[5];
  const float* W2  = (const float*)d_in[6];
  const float* as2 = (const float*)d_in[7];
  const float* ad2 = (const float*)d_in[8];
  const float* b2  = (const float*)d_in[9];
  const float* Wl  = (const float*)d_in[10];
  const float* bl  = (const float*)d_in[11];
  float* out = (float*)d_out;

  const int N  = in_sizes[0] / HIN;
  const int E  = in_sizes[1] / 2;
  const int EN = E + N;
  const int* src = ei;
  const int* dst = ei + E;

  // workspace layout (floats): XP | AGG | DEN | MAX | ASRC | ADST | EX
  float* ws = (float*)d_ws;
  const size_t NF = (size_t)N * F;
  float*    XP   = ws;
  float*    AGG  = XP + NF;
  float*    DEN  = AGG + NF;                       // N*3
  unsigned* MX   = (unsigned*)(DEN + (size_t)3 * N);
  float*    ASRC = (float*)(MX + (size_t)3 * N);
  float*    ADST = ASRC + (size_t)3 * N;
  float*    EX   = ADST + (size_t)3 * N;           // (E+N)*3

  const unsigned long long zcount = NF + (unsigned long long)6 * N; // AGG+DEN+MX
  const int zblocks = (int)((zcount + 255) / 256);
  const int eblocks = (EN + 255) / 256;
  const int bblocks = (int)((NF + 255) / 256);
  const int mtiles  = N / 16;                      // 20000/16 = 1250 (exact)
  const int gblocks = (mtiles * (F / 32) + 7) / 8; // 8 waves/block

  // ---------- layer 1 ----------
  k_zero<<<zblocks, 256, 0, stream>>>(AGG, zcount);
  k_proj1<<<N, CPH, 0, stream>>>(x, W1, as1, ad1, XP, ASRC, ADST);
  k_edge_max<<<eblocks, 256, 0, stream>>>(src, dst, E, N, ASRC, ADST, MX);
  k_edge_expsum<<<eblocks, 256, 0, stream>>>(src, dst, E, N, ASRC, ADST, MX, EX, DEN);
  k_edge_agg<<<EN, CPH, 0, stream>>>(src, dst, E, N, EX, DEN, XP, AGG);
  k_bias_rows<<<bblocks, 256, 0, stream>>>(AGG, b1, NF);      // y1 = agg1 + b1

  // ---------- layer 2 projection: XP = y1 @ W2 (fp32 WMMA) ----------
  k_gemm<<<gblocks, 256, 0, stream>>>(AGG, W2, XP, mtiles);
  k_alpha<<<N, CPH, 0, stream>>>(XP, as2, ad2, ASRC, ADST);

  // ---------- layer 2 attention ----------
  k_zero<<<zblocks, 256, 0, stream>>>(AGG, zcount);
  k_edge_max<<<eblocks, 256, 0, stream>>>(src, dst, E, N, ASRC, ADST, MX);
  k_edge_expsum<<<eblocks, 256, 0, stream>>>(src, dst, E, N, ASRC, ADST, MX, EX, DEN);
  k_edge_agg<<<EN, CPH, 0, stream>>>(src, dst, E, N, EX, DEN, XP, AGG);

  // ---------- final linear: out = (agg2 + b2) @ Wl + bl ----------
  k_final<<<N, CPH, 0, stream>>>(AGG, b2, Wl, bl, out);
}